// GCN_59081570123820
// MI455X (gfx1250) — compile-verified
//
#include <hip/hip_runtime.h>

typedef __attribute__((ext_vector_type(2))) float v2f;
typedef __attribute__((ext_vector_type(8))) float v8f;

static constexpr int kNodes  = 100000;
static constexpr int kInDim  = 128;
static constexpr int kHidDim = 128;
static constexpr int kOutDim = 64;

// ---------------- degree / symmetric norm ----------------

__global__ void k_deg_init(float* __restrict__ deg, int n) {
  int i = blockIdx.x * blockDim.x + threadIdx.x;
  if (i < n) deg[i] = 1.0f;  // self-loop contributes 1
}

__device__ __forceinline__ void atomAddF(float* p, float v) {
  unsafeAtomicAdd(p, v);  // hardware global_atomic_add_f32
}

__global__ void k_deg_count(const int* __restrict__ col, float* __restrict__ deg, int e) {
  int i = blockIdx.x * blockDim.x + threadIdx.x;
  if (i < e) atomAddF(&deg[col[i]], 1.0f);
}

__global__ void k_rsqrt_inplace(float* __restrict__ d, int n) {
  int i = blockIdx.x * blockDim.x + threadIdx.x;
  if (i < n) d[i] = rsqrtf(d[i]);
}

// ---------------- dense transform: H = (relu?)X @ W via V_WMMA_F32_16X16X4_F32 ----------------
// One wave computes a 16-row stripe across the full output width (NT tiles of 16 cols).
// VGPR layouts per CDNA5 ISA 7.12.2:
//   A (16x4 f32):  a[v] = A[lane%16][k + v + 2*(lane>=16)]
//   B (4x16 f32):  b[v] = B[k + v + 2*(lane>=16)][n0 + lane%16]
//   C/D (16x16):   c[v] = D[m0 + v + 8*(lane>=16)][n0 + lane%16]

template <int K, int NT, bool RELU>
__global__ __launch_bounds__(128) void k_gemm_wmma(const float* __restrict__ X,
                                                   const float* __restrict__ W,
                                                   float* __restrict__ H, int M) {
  constexpr int N = NT * 16;
  const int wave = blockIdx.x * 4 + (threadIdx.x >> 5);
  const int lane = threadIdx.x & 31;
  const int m0 = wave * 16;
  if (m0 >= M) return;  // uniform per wave -> EXEC stays all-ones for WMMA
  const int ml = lane & 15;
  const int hi = lane >> 4;

  v8f acc[NT];
#pragma unroll
  for (int t = 0; t < NT; ++t) acc[t] = {};

  const float* __restrict__ xrow = X + (size_t)(m0 + ml) * K;

#pragma unroll 4
  for (int k = 0; k < K; k += 4) {
    v2f a;
    a.x = xrow[k + 2 * hi + 0];
    a.y = xrow[k + 2 * hi + 1];
    if (RELU) {
      a.x = fmaxf(a.x, 0.0f);
      a.y = fmaxf(a.y, 0.0f);
    }
#pragma unroll
    for (int t = 0; t < NT; ++t) {
      v2f b;
      b.x = W[(size_t)(k + 2 * hi + 0) * N + t * 16 + ml];
      b.y = W[(size_t)(k + 2 * hi + 1) * N + t * 16 + ml];
      // 8 args: (neg_a, A, neg_b, B, c_mod, C, reuse_a, reuse_b)
      acc[t] = __builtin_amdgcn_wmma_f32_16x16x4_f32(
          false, a, false, b, (short)0, acc[t], false, false);
    }
  }

#pragma unroll
  for (int t = 0; t < NT; ++t) {
#pragma unroll
    for (int v = 0; v < 8; ++v) {
      H[(size_t)(m0 + v + 8 * hi) * N + t * 16 + ml] = acc[t][v];
    }
  }
}

// ---------------- out[i] = bias + dinv[i]^2 * H[i]  (self-loop message, initializes out) ------

template <int D>
__global__ void k_selfloop_bias(const float* __restrict__ H, const float* __restrict__ dinv,
                                const float* __restrict__ bias, float* __restrict__ out,
                                int n) {
  constexpr int C = D / 4;
  long long t = (long long)blockIdx.x * blockDim.x + threadIdx.x;
  long long total = (long long)n * C;
  if (t >= total) return;
  int i = (int)(t / C);
  int c = (int)(t % C);
  float s = dinv[i];
  float w = s * s;
  float4 h = reinterpret_cast<const float4*>(H)[t];
  float4 b = reinterpret_cast<const float4*>(bias)[c];
  float4 o;
  o.x = fmaf(w, h.x, b.x);
  o.y = fmaf(w, h.y, b.y);
  o.z = fmaf(w, h.z, b.z);
  o.w = fmaf(w, h.w, b.w);
  reinterpret_cast<float4*>(out)[t] = o;
}

// ---------------- edge scatter-add: out[col] += dinv[row]*dinv[col] * H[row] ------------------

template <int D>
__global__ void k_edge_agg(const int* __restrict__ row, const int* __restrict__ col,
                           const float* __restrict__ dinv, const float* __restrict__ H,
                           float* __restrict__ out, int e) {
  constexpr int C = D / 4;
  long long t = (long long)blockIdx.x * blockDim.x + threadIdx.x;
  long long total = (long long)e * C;
  if (t >= total) return;
  int ei = (int)(t / C);
  int c  = (int)(t % C);
  int r  = row[ei];
  int d  = col[ei];
  float norm = dinv[r] * dinv[d];
  float4 h = reinterpret_cast<const float4*>(H + (size_t)r * D)[c];
  float* o = out + (size_t)d * D + (size_t)c * 4;
  atomAddF(o + 0, norm * h.x);
  atomAddF(o + 1, norm * h.y);
  atomAddF(o + 2, norm * h.z);
  atomAddF(o + 3, norm * h.w);
}

// ---------------- launcher ----------------

extern "C" void kernel_launch(void* const* d_in, const int* in_sizes, int n_in,
                              void* d_out, int out_size, void* d_ws, size_t ws_size,
                              hipStream_t stream) {
  const float* x  = (const float*)d_in[0];
  const int*   ei = (const int*)d_in[1];   // harness contract: integer -> const int*
  const float* W1 = (const float*)d_in[2];
  const float* b1 = (const float*)d_in[3];
  const float* W2 = (const float*)d_in[4];
  const float* b2 = (const float*)d_in[5];

  const int n = kNodes;
  const int e = in_sizes[1] / 2;           // edge_index is [2, E] flat
  const int* row = ei;                     // source nodes
  const int* col = ei + e;                 // destination nodes

  // workspace layout
  char* ws = (char*)d_ws;
  float* dinv = (float*)ws;                                        // n floats
  constexpr size_t H_OFF  = 512 * 1024;                            // aligned
  constexpr size_t A1_OFF = H_OFF + (size_t)kNodes * kHidDim * 4;  // after H
  float* Hbuf = (float*)(ws + H_OFF);   // n*128 (layer1), reused as n*64 (layer2)
  float* A1   = (float*)(ws + A1_OFF);  // n*128 pre-relu layer-1 output
  float* out  = (float*)d_out;          // n*64

  const int T = 256;

  // 1) degrees (incl self-loop) -> dinv
  k_deg_init<<<(n + T - 1) / T, T, 0, stream>>>(dinv, n);
  k_deg_count<<<(e + T - 1) / T, T, 0, stream>>>(col, dinv, e);
  k_rsqrt_inplace<<<(n + T - 1) / T, T, 0, stream>>>(dinv, n);

  const int waves = (n + 15) / 16;
  const int gemmBlocks = (waves + 3) / 4;

  // 2) layer 1: H = X @ W1
  k_gemm_wmma<kInDim, kHidDim / 16, false>
      <<<gemmBlocks, 128, 0, stream>>>(x, W1, Hbuf, n);

  // 3) layer 1 aggregation: A1 = b1 + dinv^2*H  (+ edge messages)
  {
    long long sl = (long long)n * (kHidDim / 4);
    k_selfloop_bias<kHidDim>
        <<<(unsigned)((sl + T - 1) / T), T, 0, stream>>>(Hbuf, dinv, b1, A1, n);
    long long ea = (long long)e * (kHidDim / 4);
    k_edge_agg<kHidDim>
        <<<(unsigned)((ea + T - 1) / T), T, 0, stream>>>(row, col, dinv, Hbuf, A1, e);
  }

  // 4) layer 2: H2 = relu(A1) @ W2   (ReLU fused into A-loads; reuses Hbuf)
  k_gemm_wmma<kHidDim, kOutDim / 16, true>
      <<<gemmBlocks, 128, 0, stream>>>(A1, W2, Hbuf, n);

  // 5) layer 2 aggregation into d_out
  {
    long long sl = (long long)n * (kOutDim / 4);
    k_selfloop_bias<kOutDim>
        <<<(unsigned)((sl + T - 1) / T), T, 0, stream>>>(Hbuf, dinv, b2, out, n);
    long long ea = (long long)e * (kOutDim / 4);
    k_edge_agg<kOutDim>
        <<<(unsigned)((ea + T - 1) / T), T, 0, stream>>>(row, col, dinv, Hbuf, out, e);
  }
}